// NeuralSort_56126632624759
// MI455X (gfx1250) — compile-verified
//
#include <hip/hip_runtime.h>

typedef float v2f __attribute__((ext_vector_type(2)));
typedef float v4f __attribute__((ext_vector_type(4)));
typedef float v8f __attribute__((ext_vector_type(8)));

#define B_  16
#define N_  2048
#define TSTRIDE 2052   // padded LDS row stride (floats); 2052*4 % 16 == 0, kills j+8 bank aliasing

// ---------------------------------------------------------------------------
// Kernel 1: Bsum[b][i] = sum_j |s[b][i] - s[b][j]|   (16 x 2048, O(n^2) tiny)
// ---------------------------------------------------------------------------
__global__ __launch_bounds__(1024) void neuralsort_bsum_kernel(
    const float* __restrict__ scores, float* __restrict__ bsum) {
  __shared__ float sh[N_];
  const int b = blockIdx.x;
  const float* s = scores + b * N_;
  for (int i = threadIdx.x; i < N_; i += 1024) sh[i] = s[i];
  __syncthreads();
  for (int i = threadIdx.x; i < N_; i += 1024) {
    const float si = sh[i];
    float acc = 0.0f;
#pragma unroll 8
    for (int j = 0; j < N_; ++j) acc += fabsf(si - sh[j]);   // LDS broadcast reads
    bsum[b * N_ + i] = acc;
  }
}

// ---------------------------------------------------------------------------
// Kernel 2: per (b, 16-row j-block): logits via rank-2 WMMA -> LDS,
//           row softmax in LDS, non-temporal b128 streamed stores.
//   P[b, j, i] = softmax_i( scal[j]*s[b,i] - Bsum[b,i] ),  scal[j] = N-1-2j
//   Rank-2 product:  D = [scal | -1] x [ s ; Bsum ] + 0   (C = inline zero)
// ---------------------------------------------------------------------------
__global__ __launch_bounds__(256) void neuralsort_softmax_kernel(
    const float* __restrict__ scores, const float* __restrict__ bsum,
    float* __restrict__ out) {
  __shared__ float tile[16 * TSTRIDE];          // 131,328 B of the 320 KB LDS

  const int jb   = blockIdx.x;                  // 0..127  (j-block of 16 rows)
  const int b    = blockIdx.y;                  // 0..15
  const int tid  = threadIdx.x;
  const int wave = tid >> 5;                    // 0..7 (wave32)
  const int lane = tid & 31;
  const int col  = lane & 15;
  const bool hi  = lane >= 16;

  const float* s  = scores + b * N_;
  const float* bs = bsum   + b * N_;

  // A (16x4 f32), lanes 0-15: VGPR0 = K=0 col = scal[j], VGPR1 = K=1 col = -1.
  //                lanes 16-31: K=2/K=3 cols = 0.
  v2f a;
  a.x = hi ? 0.0f : (float)(N_ - 1 - 2 * (jb * 16 + lane));
  a.y = hi ? 0.0f : -1.0f;

  // Each wave fills columns [wave*256, wave*256+256) of all 16 rows: 16 tiles.
  for (int t = 0; t < 16; ++t) {
    const int i0 = wave * 256 + t * 16;

    // Unconditional loads (valid for all 32 lanes) -> no exec-mask branching.
    const float sv = s[i0 + col];
    const float bv = bs[i0 + col];

    // B (4x16 f32): VGPR0 low lanes = K=0 row = s; VGPR1 low lanes = K=1 row
    // = Bsum; upper-lane K=2/K=3 rows = 0.
    v2f bm;
    bm.x = hi ? 0.0f : sv;
    bm.y = hi ? 0.0f : bv;

    // C = 0 -> encoded as inline SRC2=0, no accumulator setup movs.
    v8f c = {};

    // D = A x B + 0  ->  v_wmma_f32_16x16x4_f32 v[d:d+7], v[a:a+1], v[b:b+1], 0
    v8f d = __builtin_amdgcn_wmma_f32_16x16x4_f32(
        /*neg_a=*/false, a, /*neg_b=*/false, bm,
        /*c_mod=*/(short)0, c, /*reuse_a=*/false, /*reuse_b=*/false);

    // C/D layout: VGPR r -> row (r + (hi?8:0)), col (lane&15)
    const int jl = hi ? 8 : 0;
#pragma unroll
    for (int r = 0; r < 8; ++r)
      tile[(jl + r) * TSTRIDE + i0 + col] = d[r];
  }
  __syncthreads();

  // Softmax: wave w exclusively owns rows {w, w+8}; no further barriers needed.
#pragma unroll
  for (int rr = 0; rr < 2; ++rr) {
    const int row = wave + rr * 8;
    v4f* trow4 = (v4f*)&tile[row * TSTRIDE];     // 16B-aligned (2052*4 % 16 == 0)

    // Pass 1: row max (b128 LDS reads, wave32 xor-shuffle reduction)
    float m = -3.402823466e38f;
    for (int i4 = lane; i4 < 512; i4 += 32) {
      const v4f v = trow4[i4];
      m = fmaxf(m, fmaxf(fmaxf(v.x, v.y), fmaxf(v.z, v.w)));
    }
#pragma unroll
    for (int off = 16; off; off >>= 1) m = fmaxf(m, __shfl_xor(m, off, 32));

    // Pass 2: exp + sum, overwrite LDS with exp values
    float ssum = 0.0f;
    for (int i4 = lane; i4 < 512; i4 += 32) {
      v4f v = trow4[i4];
      v.x = __expf(v.x - m); v.y = __expf(v.y - m);
      v.z = __expf(v.z - m); v.w = __expf(v.w - m);
      trow4[i4] = v;
      ssum += (v.x + v.y) + (v.z + v.w);
    }
#pragma unroll
    for (int off = 16; off; off >>= 1) ssum += __shfl_xor(ssum, off, 32);
    const float inv = 1.0f / ssum;

    // Pass 3: normalize + coalesced non-temporal b128 stores.
    // Output (256 MB) is write-once/never-reread; NT keeps it out of L2 (192 MB).
    v4f* orow = (v4f*)(out + ((size_t)(b * N_ + jb * 16 + row)) * N_);
    for (int i4 = lane; i4 < 512; i4 += 32) {
      v4f v = trow4[i4];
      v.x *= inv; v.y *= inv; v.z *= inv; v.w *= inv;
      __builtin_nontemporal_store(v, &orow[i4]);
    }
  }
}

// ---------------------------------------------------------------------------
extern "C" void kernel_launch(void* const* d_in, const int* in_sizes, int n_in,
                              void* d_out, int out_size, void* d_ws, size_t ws_size,
                              hipStream_t stream) {
  (void)in_sizes; (void)n_in; (void)out_size; (void)ws_size;
  const float* scores = (const float*)d_in[0];
  float* bsum = (float*)d_ws;     // 16*2048 f32 = 128 KB scratch
  float* out  = (float*)d_out;    // 16*2048*2048 f32

  neuralsort_bsum_kernel<<<dim3(B_), dim3(1024), 0, stream>>>(scores, bsum);
  neuralsort_softmax_kernel<<<dim3(N_ / 16, B_), dim3(256), 0, stream>>>(scores, bsum, out);
}